// BertEncoder_81604378624298
// MI455X (gfx1250) — compile-verified
//
#include <hip/hip_runtime.h>
#include <stdint.h>

typedef __attribute__((ext_vector_type(2))) float        v2f;
typedef __attribute__((ext_vector_type(8))) float        v8f;
typedef __attribute__((ext_vector_type(4))) unsigned int v4u;
typedef __attribute__((ext_vector_type(8))) int          v8i;
typedef __attribute__((ext_vector_type(4))) int          v4i;

#define B_ 32
#define L_ 4
#define S_ 512
#define D_ 768
#define G_ 24

#define ROWS_ 16            // rows of E_sum per TDM tile (16 KB per buffer)
#define NT_   (S_ / ROWS_)  // 32 tiles

// ---------------------------------------------------------------------------
// TDM helper: async 2-D tile load Global -> LDS (Tensor Data Mover).
// Loads `rows` lines of 256 f32, line stride D_ elements, packed densely into
// LDS at lds_byte_off. Descriptor per CDNA5 ISA ch.8 (D# groups 0/1; 2-D
// tensor so groups 2/3 are zero). Tracked by TENSORcnt.
// This toolchain exposes the 6-arg builtin (clang-23 / therock-10.0 form):
//   (uint32x4 g0, int32x8 g1, int32x4 g2, int32x4 g3, int32x8 g4, i32 cpol)
// ---------------------------------------------------------------------------
__device__ __forceinline__ void tdm_load_2d(const float* gsrc,
                                            uint32_t lds_byte_off,
                                            uint32_t rows) {
    const uint64_t ga = (uint64_t)(uintptr_t)gsrc;
    v4u g0;
    g0[0] = 1u;                                   // count=1 (valid), user mode
    g0[1] = lds_byte_off;                         // lds_addr [63:32]
    g0[2] = (uint32_t)ga;                         // global_addr [95:64]
    g0[3] = (uint32_t)((ga >> 32) & 0x01FFFFFFu)  // global_addr [120:96]
          | (2u << 30);                           // type=2 ("image") [127:126]
    v8i g1;
    g1[0] = (int)(2u << 16);                      // data_size=2 (4 bytes)
    g1[1] = (int)((256u & 0xFFFFu) << 16);        // tensor_dim0 lo -> bits[63:48]
    g1[2] = (int)((256u >> 16) | ((rows & 0xFFFFu) << 16)); // dim0 hi | tensor_dim1 lo
    g1[3] = (int)((rows >> 16) | (256u << 16));   // tensor_dim1 hi | tile_dim0=256
    g1[4] = (int)(rows & 0xFFFFu);                // tile_dim1=rows, tile_dim2=0
    g1[5] = (int)D_;                              // tensor_dim0_stride = 768 elems
    g1[6] = 0;                                    // stride hi / dim1_stride lo
    g1[7] = 0;                                    // dim1_stride hi
    v4i g2 = {0, 0, 0, 0};                        // 2-D tensor: higher dims unused
    v4i g3 = {0, 0, 0, 0};
    v8i g4 = {0, 0, 0, 0, 0, 0, 0, 0};
    __builtin_amdgcn_tensor_load_to_lds(g0, g1, g2, g3, g4, /*cpol=*/0);
}

// ---------------------------------------------------------------------------
// Kernel 1: layer sum via V_WMMA_F32_16X16X4_F32.
//   E_sum[b,s,j] = sum_{l=0..3} emb[b,l,s,j]
// A = ones(16x4) => every row of D equals the column-wise sum over K (layers).
// One block per (b,s); 8 waves x 3 tile-pairs cover D=768 (48 tiles of 16).
// B-matrix 4x16 layout (mirrors documented A 16x4): VGPR0 = K0|K2 halves,
// VGPR1 = K1|K3; any K permutation is harmless since all layers get weight 1.
// Pairwise stores: lanes 0-15 take tile t from c0[0] (row M=0), lanes 16-31
// take tile t+1 from c1[0] (row M=8, identical values) -> one contiguous
// full-wave 128B store, no EXEC masking.
// ---------------------------------------------------------------------------
__global__ void layer_sum_wmma(const float* __restrict__ emb,
                               float* __restrict__ esum) {
    const int bs   = blockIdx.x;      // 0 .. B*S-1
    const int b    = bs / S_;
    const int s    = bs % S_;
    const int wave = threadIdx.x >> 5;
    const int lane = threadIdx.x & 31;
    const int half = lane >> 4;
    const int ln   = lane & 15;

    const size_t base    = ((size_t)b * L_ * S_ + s) * D_;  // emb[b,0,s,0]
    const size_t lstride = (size_t)S_ * D_;
    const size_t row     = ((size_t)b * S_ + s) * D_;       // esum[b,s,0]
    const int    l0      = half * 2;  // lanes 0-15: layers {0,1}; 16-31: {2,3}

    const v2f a = {1.0f, 1.0f};       // ones A-matrix (16x4)

    for (int p = 0; p < 3; ++p) {
        const int j0 = (wave * 6 + p * 2) * 16;
        v2f bm0, bm1;
        bm0[0] = emb[base + (size_t)(l0 + 0) * lstride + j0 + ln];
        bm0[1] = emb[base + (size_t)(l0 + 1) * lstride + j0 + ln];
        bm1[0] = emb[base + (size_t)(l0 + 0) * lstride + j0 + 16 + ln];
        bm1[1] = emb[base + (size_t)(l0 + 1) * lstride + j0 + 16 + ln];
        v8f c0 = {}, c1 = {};
        c0 = __builtin_amdgcn_wmma_f32_16x16x4_f32(false, a, false, bm0,
                                                   (short)0, c0, false, false);
        c1 = __builtin_amdgcn_wmma_f32_16x16x4_f32(false, a, false, bm1,
                                                   (short)0, c1, false, false);
        const float r = (half == 0) ? c0[0] : c1[0];
        esum[row + j0 + lane] = r;    // contiguous across the wave
    }
}

// ---------------------------------------------------------------------------
// Kernel 2: fused deterministic scatter pass over tokens, TDM-staged.
// One block per (batch b, 256-column chunk); each thread owns one column.
// E_sum tiles (ROWS_ x 256 f32) stream Global->LDS via the Tensor Data Mover,
// double-buffered: wave 0 issues tile k+1, waits TENSORcnt<=1 (k complete),
// barrier publishes the tile, all 8 waves consume it. Both id streams are
// monotone, so word/segment sums are contiguous runs: register-accumulate,
// flush with a plain store on id change (deterministic, no atomics).
// ---------------------------------------------------------------------------
__global__ void fused_scatter(const float* __restrict__ esum,
                              const int*   __restrict__ tgids,
                              const int*   __restrict__ segids,
                              float* __restrict__ out_word,   // [B,S,D]
                              float* __restrict__ out_sent,   // [B,D]
                              float* __restrict__ out_seg) {  // [B,G,D]
    __shared__ float tile[2][ROWS_ * 256];
    __shared__ int   sh_tg[S_];   // word id per token
    __shared__ int   sh_sg[S_];   // segment id per token (composed)

    const int chunks = D_ / 256;              // 3
    const int b      = blockIdx.x / chunks;
    const int chunk  = blockIdx.x % chunks;
    const int col    = chunk * 256 + threadIdx.x;
    const float* ebase = esum + (size_t)b * S_ * D_ + chunk * 256;

    // Kick off tile 0 DMA immediately (only wave 0 issues; TDM ignores EXEC).
    if (threadIdx.x == 0)
        tdm_load_2d(ebase, (uint32_t)(uintptr_t)&tile[0][0], ROWS_);

    for (int i = threadIdx.x; i < S_; i += 256)
        sh_tg[i] = tgids[b * S_ + i];
    __syncthreads();
    for (int i = threadIdx.x; i < S_; i += 256)
        sh_sg[i] = segids[b * S_ + sh_tg[i]];
    __syncthreads();

    float acc_w = 0.0f, acc_g = 0.0f, acc_s = 0.0f;

    for (int k = 0; k < NT_; ++k) {
        if (threadIdx.x == 0) {
            if (k + 1 < NT_) {
                // prefetch next tile into the ping-pong buffer, then wait for
                // tile k (TDM completes in order within a wave)
                tdm_load_2d(ebase + (size_t)(k + 1) * ROWS_ * D_,
                            (uint32_t)(uintptr_t)&tile[(k + 1) & 1][0], ROWS_);
                __builtin_amdgcn_s_wait_tensorcnt(1);
            } else {
                __builtin_amdgcn_s_wait_tensorcnt(0);
            }
        }
        __syncthreads();                       // tile k visible to all waves

        const float* tp = &tile[k & 1][0];
        for (int r = 0; r < ROWS_; ++r) {
            const int s = k * ROWS_ + r;
            const float v = tp[r * 256 + threadIdx.x];
            acc_w += v; acc_g += v; acc_s += v;

            const bool last = (s == S_ - 1);
            if (last || sh_tg[s + 1] != sh_tg[s]) {
                out_word[((size_t)b * S_ + sh_tg[s]) * D_ + col] = acc_w;
                acc_w = 0.0f;
            }
            if (last || sh_sg[s + 1] != sh_sg[s]) {
                out_seg[((size_t)b * G_ + sh_sg[s]) * D_ + col] = acc_g;
                acc_g = 0.0f;
            }
        }
        __syncthreads();                       // done reading before buffer reuse
    }
    out_sent[(size_t)b * D_ + col] = acc_s * (1.0f / (float)S_);
}

// ---------------------------------------------------------------------------
// Kernel 3: seg_mask[b,g] = (g >= segids[b, S-1] + 1) ? 1 : 0  (ids sorted)
// ---------------------------------------------------------------------------
__global__ void seg_mask_kernel(const int* __restrict__ segids,
                                float* __restrict__ out_mask) {
    const int b = blockIdx.x;
    const int g = threadIdx.x;
    if (g < G_) {
        const int seg_num = segids[b * S_ + (S_ - 1)] + 1;
        out_mask[b * G_ + g] = (g >= seg_num) ? 1.0f : 0.0f;
    }
}

// ---------------------------------------------------------------------------
// Kernel 0: zero the scatter-target output regions.
// ---------------------------------------------------------------------------
__global__ void zero_f32(float* __restrict__ p, size_t n) {
    const size_t i = (size_t)blockIdx.x * blockDim.x + threadIdx.x;
    if (i < n) p[i] = 0.0f;
}

extern "C" void kernel_launch(void* const* d_in, const int* in_sizes, int n_in,
                              void* d_out, int out_size, void* d_ws, size_t ws_size,
                              hipStream_t stream) {
    const float* emb = (const float*)d_in[0];   // [B,L,S,D] f32
    const int*   tg  = (const int*)d_in[1];     // [B,S] i32 (sorted per row)
    const int*   sg  = (const int*)d_in[2];     // [B,S] i32 (sorted per row)
    float* out  = (float*)d_out;
    float* esum = (float*)d_ws;                 // [B,S,D] f32 = 50.3 MB scratch

    // Output layout (reference tuple return order, concatenated flat):
    const size_t off_word = 0;                                 // [B,S,D]
    const size_t off_sent = (size_t)B_ * S_ * D_;              // [B,D]
    const size_t off_seg  = off_sent + (size_t)B_ * D_;        // [B,G,D]
    const size_t off_mask = off_seg + (size_t)B_ * G_ * D_;    // [B,G]

    // 0) zero word/sent/seg regions (scatter targets; sent overwritten anyway)
    const size_t nz = off_mask;
    zero_f32<<<(unsigned)((nz + 255) / 256), 256, 0, stream>>>(out, nz);

    // 1) layer sum via f32 WMMA -> E_sum workspace
    layer_sum_wmma<<<B_ * S_, 256, 0, stream>>>(emb, esum);

    // 2) fused word/segment/sentence pass (TDM double-buffered, no atomics)
    fused_scatter<<<B_ * (D_ / 256), 256, 0, stream>>>(
        esum, tg, sg, out + off_word, out + off_sent, out + off_seg);

    // 3) segment padding mask
    seg_mask_kernel<<<B_, 32, 0, stream>>>(sg, out + off_mask);
}